// LearnedColorPool_56650618634977
// MI455X (gfx1250) — compile-verified
//
#include <hip/hip_runtime.h>
#include <hip/hip_bf16.h>

typedef __attribute__((ext_vector_type(2))) float v2f;
typedef __attribute__((ext_vector_type(8))) float v8f;

// Problem constants from the reference: B,N,C,D = 16,4096,1024,8 ; k = N/2
constexpr int Bsz = 16;
constexpr int Nn  = 4096;
constexpr int Cc  = 1024;
constexpr int Dd  = 8;
constexpr int Kk  = Nn / 2;

constexpr int XS_STRIDE = Cc + 8;   // pad LDS rows to dodge bank conflicts

// ---------------------------------------------------------------------------
// Kernel 1: fused attn[b,n] = x_n^T W x_n + b.x_n  via V_WMMA_F32_16X16X4_F32,
// plus scale[b,n] = 1 + |tanh(attn)| for the pooling pass.
// One block = one 16-row M-tile; 8 waves, each owns 8 N-tiles of W.
// ---------------------------------------------------------------------------
__global__ __launch_bounds__(256)
void attn_gemm_kernel(const float* __restrict__ x,      // [B*N, C]
                      const float* __restrict__ W,      // [C, C] row-major
                      const float* __restrict__ bvec,   // [C]
                      float* __restrict__ attn_g,       // [B*N]
                      float* __restrict__ scale_g)      // [B*N]
{
    __shared__ float xs[16 * XS_STRIDE];
    __shared__ float attn_s[16];

    const int  tid = threadIdx.x;
    const long m0  = (long)blockIdx.x * 16;      // first row of this M-tile

    if (tid < 16) attn_s[tid] = 0.0f;
    __syncthreads();

    // Stage 16x1024 x-tile into LDS; iteration r covers exactly row r
    // (256 threads * float4 = 1024 elems). Fold b.x into attn on the way in.
    const float* xrow_base = x + m0 * Cc;
    const int c0 = tid * 4;
    for (int r = 0; r < 16; ++r) {
        const float4 v  = *reinterpret_cast<const float4*>(xrow_base + (long)r * Cc + c0);
        const float4 bv = *reinterpret_cast<const float4*>(bvec + c0);
        xs[r * XS_STRIDE + c0 + 0] = v.x;
        xs[r * XS_STRIDE + c0 + 1] = v.y;
        xs[r * XS_STRIDE + c0 + 2] = v.z;
        xs[r * XS_STRIDE + c0 + 3] = v.w;
        const float p = v.x * bv.x + v.y * bv.y + v.z * bv.z + v.w * bv.w;
        atomicAdd(&attn_s[r], p);                // ds_add_f32
    }
    __syncthreads();

    const int wave = tid >> 5;
    const int lane = tid & 31;
    const int half = lane >> 4;                  // 0: lanes 0-15, 1: lanes 16-31
    const int lrow = lane & 15;                  // M (for A) / N (for B) position

    for (int nt = wave; nt < Cc / 16; nt += 8) {
        const int col = nt * 16 + lrow;
        v8f acc = {0.f, 0.f, 0.f, 0.f, 0.f, 0.f, 0.f, 0.f};

        #pragma unroll 4
        for (int k = 0; k < Cc; k += 4) {
            const int ka = k + 2 * half;         // A: lane holds K = 2*half + {0,1}
            v2f a, bb;
            a.x  = xs[lrow * XS_STRIDE + ka];
            a.y  = xs[lrow * XS_STRIDE + ka + 1];
            bb.x = W[(long)ka * Cc + col];       // B: row K=ka      , col = n0+lane%16
            bb.y = W[((long)ka + 1) * Cc + col]; // B: row K=ka+1
            acc = __builtin_amdgcn_wmma_f32_16x16x4_f32(
                      false, a, false, bb, (short)0, acc, false, false);
        }

        // acc[r] = emb[m = r + 8*half][col]; reduce emb*x into per-row attn.
        #pragma unroll
        for (int r = 0; r < 8; ++r) {
            const int m = r + 8 * half;
            atomicAdd(&attn_s[m], acc[r] * xs[m * XS_STRIDE + col]);
        }
    }
    __syncthreads();

    if (tid < 16) {
        const float av = attn_s[tid];
        attn_g[m0 + tid]  = av;
        scale_g[m0 + tid] = 1.0f + fabsf(tanhf(av));   // relu(x*a+x) = (1+a)*relu(x)
    }
}

// ---------------------------------------------------------------------------
// Kernel 2: per-batch top-k (k = N/2) via full bitonic sort of (val, idx)
// pairs in LDS. Descending by value, ascending index tiebreak (jax top_k order).
// ---------------------------------------------------------------------------
__global__ __launch_bounds__(1024)
void topk_kernel(const float* __restrict__ attn_g,   // [B*N]
                 int*   __restrict__ centers_g,      // [B*k] (workspace, int)
                 float* __restrict__ centers_out)    // [B*k] (d_out tail, as float)
{
    __shared__ float vals[Nn];
    __shared__ int   idxs[Nn];

    const int b   = blockIdx.x;
    const int tid = threadIdx.x;

    for (int i = tid; i < Nn; i += blockDim.x) {
        vals[i] = attn_g[(long)b * Nn + i];
        idxs[i] = i;
    }
    __syncthreads();

    for (int size = 2; size <= Nn; size <<= 1) {
        for (int stride = size >> 1; stride > 0; stride >>= 1) {
            for (int t = tid; t < Nn / 2; t += blockDim.x) {
                const int i = 2 * t - (t & (stride - 1));
                const int j = i + stride;
                const bool descending = ((i & size) == 0);
                const float vi = vals[i], vj = vals[j];
                const int   ii = idxs[i], ij = idxs[j];
                const bool gi = (vi > vj) || (vi == vj && ii < ij);
                if (gi != descending) {
                    vals[i] = vj; vals[j] = vi;
                    idxs[i] = ij; idxs[j] = ii;
                }
            }
            __syncthreads();
        }
    }

    for (int i = tid; i < Kk; i += blockDim.x) {
        const int c = idxs[i];
        centers_g[b * Kk + i] = c;
        centers_out[(long)b * Kk + i] = (float)c;
    }
}

// ---------------------------------------------------------------------------
// Kernel 3: out[b,i,:] = max over {center} u neigh(center) of (1+a_j)*relu(x_j)
// One block per (center, batch); attended tensor never materialized.
// ---------------------------------------------------------------------------
__global__ __launch_bounds__(256)
void pool_kernel(const float* __restrict__ x,         // [B*N, C]
                 const int*   __restrict__ neigh,     // [B, N, D]
                 const int*   __restrict__ centers_g, // [B*k]
                 const float* __restrict__ scale_g,   // [B*N]
                 float* __restrict__ out)             // [B*k, C]
{
    __shared__ int   rows_s[Dd + 1];
    __shared__ float sc_s[Dd + 1];

    const int i   = blockIdx.x;      // center slot 0..k-1
    const int b   = blockIdx.y;      // batch
    const int tid = threadIdx.x;

    if (tid < Dd + 1) {
        const int center = centers_g[b * Kk + i];
        int row = center;
        if (tid > 0) row = neigh[((long)b * Nn + center) * Dd + (tid - 1)];
        rows_s[tid] = row;
        sc_s[tid]   = scale_g[(long)b * Nn + row];
    }
    __syncthreads();

    const int c0 = tid * 4;
    float4 acc = make_float4(0.f, 0.f, 0.f, 0.f);   // attended >= 0 always
    #pragma unroll
    for (int j = 0; j < Dd + 1; ++j) {
        const float  s = sc_s[j];
        const float4 v = *reinterpret_cast<const float4*>(
                             x + ((long)b * Nn + rows_s[j]) * Cc + c0);
        acc.x = fmaxf(acc.x, s * fmaxf(v.x, 0.f));
        acc.y = fmaxf(acc.y, s * fmaxf(v.y, 0.f));
        acc.z = fmaxf(acc.z, s * fmaxf(v.z, 0.f));
        acc.w = fmaxf(acc.w, s * fmaxf(v.w, 0.f));
    }
    *reinterpret_cast<float4*>(out + ((long)b * Kk + i) * Cc + c0) = acc;
}

// ---------------------------------------------------------------------------
extern "C" void kernel_launch(void* const* d_in, const int* in_sizes, int n_in,
                              void* d_out, int out_size, void* d_ws, size_t ws_size,
                              hipStream_t stream) {
    (void)in_sizes; (void)n_in; (void)out_size; (void)ws_size;

    const float* x     = (const float*)d_in[0];   // [B,N,C] f32
    const float* W     = (const float*)d_in[1];   // [C,C]   f32
    const float* bvec  = (const float*)d_in[2];   // [C]     f32
    const int*   neigh = (const int*)d_in[3];     // [B,N,D] i32

    float* out         = (float*)d_out;                      // [B,k,C]
    float* centers_out = out + (long)Bsz * Kk * Cc;          // [B,k] as float

    // Workspace layout: attn | scale | centers(int)  (640 KB total)
    float* attn_ws   = (float*)d_ws;
    float* scale_ws  = attn_ws + (long)Bsz * Nn;
    int*   centers_ws = (int*)(scale_ws + (long)Bsz * Nn);

    attn_gemm_kernel<<<dim3(Bsz * Nn / 16), dim3(256), 0, stream>>>(
        x, W, bvec, attn_ws, scale_ws);

    topk_kernel<<<dim3(Bsz), dim3(1024), 0, stream>>>(
        attn_ws, centers_ws, centers_out);

    pool_kernel<<<dim3(Kk, Bsz), dim3(256), 0, stream>>>(
        x, neigh, centers_ws, scale_ws, out);
}